// Prior_SPDE_82300163326674
// MI455X (gfx1250) — compile-verified
//
#include <hip/hip_runtime.h>

// ---------------------------------------------------------------------------
// Prior_SPDE: Q = block-tridiagonal assembly of MM = (I+A)^2 and -(I+A)
// Output: (4, 5120, 5120) f32 = 419 MB  -> HBM-write bound (~18 us @ 23.3TB/s).
// Strategy:
//   1) coeff_kernel : materialized 3x3 stencil table per (b,t,node)
//                     (16 floats/record, slots 9..15 = 0) -> 1.3 MB in d_ws
//   2) zero_kernel  : non-temporal float4 zero-fill of all of Q (Q > L2)
//   3) tile_kernel  : one wave per 16x16 band tile.
//        off-diag blocks: -M[j] stencil tiles (|dTile|<=3), branchless gather
//        diag blocks:     MM[i] tiles (|dTile|<=5) via banded f32 GEMM with
//                         v_wmma_f32_16x16x4_f32, K limited to the stencil
//                         band (|k-row|<=33 and |k-col|<=33).
// All fragment-element selection is branchless (index clamp into a stored-0
// slot), so the K loop has no EXEC manipulation around the WMMA.
// ---------------------------------------------------------------------------

typedef float v2f __attribute__((ext_vector_type(2)));
typedef float v8f __attribute__((ext_vector_type(8)));
typedef float f4  __attribute__((ext_vector_type(4)));

#define N_TIME 5
#define NNODE  1024            // 32 x 32 grid
#define LQ     5120            // N_TIME * NNODE
#define TPB    64              // 16x16 tiles per row of an NxN block (1024/16)
#define REC    16              // floats per stencil record

// M[r][c] via one branchless gather from row r's 3x3 stencil table.
// tbl[(dy+1)*3 + (dx+1)] holds the stencil value; tbl[9..15] == 0.
__device__ __forceinline__ float m_entry(const float* __restrict__ tbl, int r, int c) {
    int dy = (c >> 5) - (r >> 5) + 1;   // 0..2 if in stencil
    int dx = (c & 31) - (r & 31) + 1;
    bool ok = ((unsigned)dy <= 2u) & ((unsigned)dx <= 2u);
    int idx = ok ? (dy * 3 + dx) : 9;   // slot 9 stores 0.0f
    return tbl[idx];
}

__global__ void coeff_kernel(const float* __restrict__ kappa,
                             const float* __restrict__ m,
                             const float* __restrict__ H,
                             float* __restrict__ cf) {
    int id = blockIdx.x * blockDim.x + threadIdx.x;   // 4*5*1024 = 20480
    if (id >= 4 * N_TIME * NNODE) return;
    int node = id % NNODE;
    int bt   = id / NNODE;
    int t    = bt % N_TIME;
    int b    = bt / N_TIME;

    float kap = kappa[((size_t)(b * NNODE + node)) * N_TIME + t];
    float mx  = m[(((size_t)(b * 2 + 0)) * NNODE + node) * N_TIME + t];
    float my  = m[(((size_t)(b * 2 + 1)) * NNODE + node) * N_TIME + t];
    float H11 = H[((((size_t)(b * 2 + 0)) * 2 + 0) * NNODE + node) * N_TIME + t];
    float H12 = H[((((size_t)(b * 2 + 0)) * 2 + 1) * NNODE + node) * N_TIME + t];
    float H21 = H[((((size_t)(b * 2 + 1)) * 2 + 0) * NNODE + node) * N_TIME + t];
    float H22 = H[((((size_t)(b * 2 + 1)) * 2 + 1) * NNODE + node) * N_TIME + t];

    float c0  = 1.0f + kap * kap + 2.0f * H11 + 2.0f * H22;  // M diag = c0
    float cpx =  0.5f * mx - H11;
    float cmx = -0.5f * mx - H11;
    float cpy =  0.5f * my - H22;
    float cmy = -0.5f * my - H22;
    float cc  =  0.25f * (H12 + H21);

    f4* o = (f4*)(cf + (size_t)id * REC);
    // (dy,dx): idx = (dy+1)*3 + (dx+1)
    f4 r0 = { -cc, cmy,  cc, cmx };    // (-1,-1) (-1,0) (-1,1) (0,-1)
    f4 r1 = {  c0, cpx,  cc, cpy };    // (0,0)   (0,1)  (1,-1) (1,0)
    f4 r2 = { -cc, 0.f, 0.f, 0.f };    // (1,1)   pad0   pad0   pad0
    f4 r3 = { 0.f, 0.f, 0.f, 0.f };
    o[0] = r0; o[1] = r1; o[2] = r2; o[3] = r3;
}

__global__ void zero_kernel(f4* __restrict__ q4, long n4) {
    long i      = (long)blockIdx.x * blockDim.x + threadIdx.x;
    long stride = (long)gridDim.x * blockDim.x;
    f4 z = {0.0f, 0.0f, 0.0f, 0.0f};
    for (; i < n4; i += stride)
        __builtin_nontemporal_store(z, q4 + i);   // Q >> L2: stream past caches
}

// One wave per 16x16 tile. Wave-id decode is uniform per wave, so EXEC is
// all-ones at every WMMA (required by the ISA).
__global__ void tile_kernel(const float* __restrict__ cf, float* __restrict__ q) {
    const int wave = blockIdx.x * (blockDim.x >> 5) + (threadIdx.x >> 5);
    const int lane = threadIdx.x & 31;

    // 25 jobs per tile-row: 11 diag (dt in [-5,5]), 7 left, 7 right (dt in [-3,3])
    int b    = wave / (N_TIME * TPB * 25);
    int rem  = wave % (N_TIME * TPB * 25);
    int i    = rem / (TPB * 25);
    int rem2 = rem % (TPB * 25);
    int tr   = rem2 / 25;
    int s    = rem2 % 25;

    int j, dt;
    bool diag;
    if (s < 11)      { diag = true;  j = i;     dt = s - 5;  }
    else if (s < 18) { diag = false; j = i - 1; dt = s - 14; }
    else             { diag = false; j = i + 1; dt = s - 21; }
    if (j < 0 || j >= N_TIME) return;          // uniform
    int tc = tr + dt;
    if (tc < 0 || tc >= TPB) return;           // uniform

    const int r0 = tr * 16, c0 = tc * 16;
    const float* cfbt = cf + ((size_t)(b * N_TIME + j) * NNODE) * REC; // time = col block
    float* qb = q + (size_t)b * LQ * LQ;
    const size_t grow = (size_t)i * NNODE + r0;
    const size_t gcol = (size_t)j * NNODE + c0;
    const int hi  = lane >> 4;     // half-wave select
    const int l16 = lane & 15;     // tile column (B/C/D), tile row (A)

    if (!diag) {
        // off-diagonal block: -M[j] tile, one branchless gather per element
        #pragma unroll
        for (int kk = 0; kk < 8; ++kk) {
            int row = kk + 8 * hi;
            const float* tbl = cfbt + (size_t)(r0 + row) * REC;
            float v = m_entry(tbl, r0 + row, c0 + l16);
            qb[(grow + row) * LQ + gcol + l16] = -v;
        }
        return;
    }

    // diagonal block: C = sum_k M[R,k] * M[k,C] over the stencil band.
    // M[r,k] != 0 only for |k-r| <= 33  ->  k in [max(r0,c0)-33, min(r0,c0)+48]
    int kmin = (r0 > c0 ? r0 : c0) - 33; if (kmin < 0) kmin = 0;
    int kmax = (r0 < c0 ? r0 : c0) + 48; if (kmax > NNODE - 1) kmax = NNODE - 1;

    v8f acc = {};
    const int rowA = r0 + l16;                       // A-frag row = lane%16
    const float* tblA = cfbt + (size_t)rowA * REC;   // reused across the K loop
    const int colB = c0 + l16;

    for (int k = (kmin & ~3); k <= kmax; k += 4) {
        int ka = k + 2 * hi;                         // this lane's K pair
        int ka0 = (ka     < NNODE) ? ka     : 0;     // clamp; value forced 0 below
        int ka1 = (ka + 1 < NNODE) ? ka + 1 : 0;
        v2f a, bm;
        // A 16x4: VGPR0 holds K = 2*hi, VGPR1 holds K = 2*hi+1
        a.x = (ka     < NNODE) ? m_entry(tblA, rowA, ka)     : 0.0f;
        a.y = (ka + 1 < NNODE) ? m_entry(tblA, rowA, ka + 1) : 0.0f;
        // B 4x16: same K striping; row of B = k index, col = lane%16
        bm.x = m_entry(cfbt + (size_t)ka0 * REC, ka0, colB);
        bm.y = m_entry(cfbt + (size_t)ka1 * REC, ka1, colB);
        // (if ka >= NNODE the a element is already 0, so bm garbage-row values
        //  from row 0 contribute a.x*bm.x = 0; and ka0=0 is a valid record)
        acc = __builtin_amdgcn_wmma_f32_16x16x4_f32(
            /*neg_a=*/false, a, /*neg_b=*/false, bm,
            /*c_mod=*/(short)0, acc, /*reuse_a=*/false, /*reuse_b=*/false);
    }

    // interior time blocks get MM + I
    const bool interior = (i >= 1 && i <= N_TIME - 2);
    #pragma unroll
    for (int kk = 0; kk < 8; ++kk) {
        int row = kk + 8 * hi;
        float v = acc[kk];
        if (interior && (r0 + row) == (c0 + l16)) v += 1.0f;
        qb[(grow + row) * LQ + gcol + l16] = v;
    }
}

extern "C" void kernel_launch(void* const* d_in, const int* in_sizes, int n_in,
                              void* d_out, int out_size, void* d_ws, size_t ws_size,
                              hipStream_t stream) {
    const float* kappa = (const float*)d_in[0];
    const float* m     = (const float*)d_in[1];
    const float* H     = (const float*)d_in[2];
    float* q  = (float*)d_out;
    float* cf = (float*)d_ws;    // 4*5*1024*16 floats = 1.31 MB

    // 1) stencil tables (tiny, L2/L0-resident afterwards)
    coeff_kernel<<<(4 * N_TIME * NNODE + 255) / 256, 256, 0, stream>>>(kappa, m, H, cf);

    // 2) stream-zero all of Q with NT float4 stores (bandwidth floor: 419 MB)
    long n4 = (long)out_size / 4;
    zero_kernel<<<8192, 256, 0, stream>>>((f4*)q, n4);

    // 3) band tiles: 4 batches * 5 block-rows * 64 tile-rows * 25 jobs = 32000 waves
    int waves  = 4 * N_TIME * TPB * 25;
    int blocks = waves / 8;                    // 256 threads = 8 waves/block
    tile_kernel<<<blocks, 256, 0, stream>>>(cf, q);
}